// VectorQuantizer_46188078301438
// MI455X (gfx1250) — compile-verified
//
#include <hip/hip_runtime.h>
#include <math.h>

// ---------------------------------------------------------------------------
// VQ-VAE vector quantizer for MI455X (gfx1250, wave32, WMMA).
//
// Outputs concatenated in d_out (float):
//   [0)          zq_out  : 32*256*32*32 = 8388608   (NCHW, == zq numerically)
//   [8388608)    loss    : 1
//   [8388609)    perplex : 1
//   [8388610)    nearest : 32768*1024   = 33554432  (one-hot)
//   [41943042)   idx     : 32768                    (as float)
// ---------------------------------------------------------------------------

#define EMB_DIM   256
#define N_EMB     1024
#define N_ROWS    32768        // 32 batches * 32 * 32
#define BETA      0.25f

#define ROWS_PER_BLK 128
#define COLS_PER_BLK 64
#define SD           260       // padded LDS row stride in floats (bank-conflict free frags)

#define OFF_ZQ    0ull
#define OFF_LOSS  8388608ull
#define OFF_PERP  8388609ull
#define OFF_NEAR  8388610ull
#define OFF_IDX   41943042ull

// workspace layout (bytes)
#define WS_IDX    0            // int   [32768]
#define WS_HIST   131072       // int   [1024]
#define WS_PART   135168       // float [4096]
#define WS_EB2    151552       // float [1024]
#define NPART     4096

typedef float v2f __attribute__((ext_vector_type(2)));
typedef float v8f __attribute__((ext_vector_type(8)));

__device__ __forceinline__ unsigned long long packKey(float v, int idx) {
  unsigned u = __float_as_uint(v);
  u ^= (u & 0x80000000u) ? 0xFFFFFFFFu : 0x80000000u;   // monotonic unsigned order
  return ((unsigned long long)u << 32) | (unsigned)idx; // ties -> smaller index
}

// ---------------------------------------------------------------------------
// Kernel 0: zero the one-hot output region (float2 stores; region is 8B-aligned)
//           and the histogram.
// ---------------------------------------------------------------------------
__global__ __launch_bounds__(256) void vq_init_kernel(float* __restrict__ out,
                                                      int* __restrict__ hist) {
  const unsigned t = threadIdx.x;
  v2f* nr = (v2f*)(out + OFF_NEAR);
  const unsigned base = blockIdx.x * 8192u + t;          // float2 units
  const v2f zz = {0.0f, 0.0f};
#pragma unroll
  for (int j = 0; j < 32; ++j) nr[base + j * 256u] = zz;
  if (blockIdx.x == 0) {
    for (int i = t; i < N_EMB; i += 256) hist[i] = 0;
  }
}

// ---------------------------------------------------------------------------
// Kernel 0b: eb2[n] = ||emb[n]||^2  (tiny: 1 MB read once)
// ---------------------------------------------------------------------------
__global__ __launch_bounds__(256) void vq_eb2_kernel(const float* __restrict__ emb,
                                                     float* __restrict__ eb2) {
  const int n = blockIdx.x * 256 + threadIdx.x;   // 0..1023
  float acc = 0.0f;
  const float* e = emb + n * EMB_DIM;
#pragma unroll 8
  for (int c = 0; c < EMB_DIM; ++c) { const float v = e[c]; acc += v * v; }
  eb2[n] = acc;
}

// ---------------------------------------------------------------------------
// Kernel 1: distances argmin via V_WMMA_F32_16X16X4_F32.
// Block: 256 threads (8 waves) = 4 row-groups x 2 col-groups; wave tile 32x32.
// argmin_n dist(m,n) == argmin_n ( ||e_n||^2 - 2 * z_m . e_n )   (||z||^2 const)
// Per-lane running min across all col-blocks; single cross-lane u64-key
// reduction at the end; one-hot scatter fused into the tail.
// ---------------------------------------------------------------------------
__global__ __launch_bounds__(256) void vq_argmin_kernel(const float* __restrict__ z,
                                                        const float* __restrict__ emb,
                                                        const float* __restrict__ eb2,
                                                        float* __restrict__ out,
                                                        int* __restrict__ g_idx,
                                                        int* __restrict__ g_hist) {
  __shared__ float sZ[ROWS_PER_BLK * SD];          // 133120 B
  __shared__ float sE[COLS_PER_BLK * SD];          //  66560 B
  __shared__ unsigned long long sKey[ROWS_PER_BLK][2];

  const int t    = threadIdx.x;
  const int lane = t & 31;
  const int wave = t >> 5;       // 0..7
  const int rg   = wave >> 1;    // row group 0..3
  const int cg   = wave & 1;     // col group 0..1
  const int q    = lane & 15;
  const int half = lane >> 4;

  const int blk    = blockIdx.x;             // 0..255, 128 rows each
  const int b      = blk >> 3;               // batch (1024 rows per batch / 128)
  const int hwbase = (blk & 7) * ROWS_PER_BLK;
  const long long zoff = (long long)b * (EMB_DIM * 1024) + hwbase;

  // Stage z tile transposed: sZ[i][c] = z[b, c, hwbase+i]  (coalesced global reads)
  for (int s = t; s < ROWS_PER_BLK * EMB_DIM; s += 256) {
    const int c = s >> 7;        // 0..255
    const int i = s & 127;       // 0..127
    sZ[i * SD + c] = z[zoff + (long long)c * 1024 + i];
  }

  const int rbase = rg * 32;
  const int lc    = cg * 32;
  const float* zb0 = &sZ[(rbase + q) * SD + (half << 1)];
  const float* zb1 = zb0 + 16 * SD;

  float runV0[8], runV1[8];
  int   runI0[8], runI1[8];
#pragma unroll
  for (int r = 0; r < 8; ++r) {
    runV0[r] = 3.4e38f; runV1[r] = 3.4e38f; runI0[r] = 0; runI1[r] = 0;
  }

  for (int cblk = 0; cblk < N_EMB / COLS_PER_BLK; ++cblk) {
    const int colbase = cblk * COLS_PER_BLK;
    __syncthreads();                               // previous sE readers done / sZ staged
    for (int s = t; s < COLS_PER_BLK * EMB_DIM; s += 256) {
      const int j = s >> 8;                        // col within tile
      const int c = s & 255;                       // dim
      sE[j * SD + c] = emb[(colbase + j) * EMB_DIM + c];
    }
    __syncthreads();

    const float* eb0 = &sE[(lc + q) * SD + (half << 1)];
    const float* eb1 = eb0 + 16 * SD;

    v8f c00 = {}, c01 = {}, c10 = {}, c11 = {};
#pragma unroll 4
    for (int kc = 0; kc < EMB_DIM / 4; ++kc) {
      const int ko = kc * 4;
      const v2f a0 = *(const v2f*)(zb0 + ko);
      const v2f a1 = *(const v2f*)(zb1 + ko);
      const v2f b0 = *(const v2f*)(eb0 + ko);
      const v2f b1 = *(const v2f*)(eb1 + ko);
      c00 = __builtin_amdgcn_wmma_f32_16x16x4_f32(false, a0, false, b0, (short)0, c00, false, false);
      c01 = __builtin_amdgcn_wmma_f32_16x16x4_f32(false, a0, false, b1, (short)0, c01, false, false);
      c10 = __builtin_amdgcn_wmma_f32_16x16x4_f32(false, a1, false, b0, (short)0, c10, false, false);
      c11 = __builtin_amdgcn_wmma_f32_16x16x4_f32(false, a1, false, b1, (short)0, c11, false, false);
    }

    const int   n0   = colbase + lc + q;
    const int   n1   = n0 + 16;
    const float eb2a = eb2[n0];                    // coalesced, L2-hot
    const float eb2b = eb2[n1];

    // Branchless per-lane running-min update (newer indices are always larger,
    // so strict '<' preserves first-occurrence tie-breaking).
#pragma unroll
    for (int r = 0; r < 8; ++r) {
      {
        const float v0 = eb2a - 2.0f * c00[r];
        const float v1 = eb2b - 2.0f * c01[r];
        const bool  s1 = v1 < v0;
        const float v  = s1 ? v1 : v0;
        const int   ci = s1 ? n1 : n0;
        if (v < runV0[r]) { runV0[r] = v; runI0[r] = ci; }
      }
      {
        const float v0 = eb2a - 2.0f * c10[r];
        const float v1 = eb2b - 2.0f * c11[r];
        const bool  s1 = v1 < v0;
        const float v  = s1 ? v1 : v0;
        const int   ci = s1 ? n1 : n0;
        if (v < runV1[r]) { runV1[r] = v; runI1[r] = ci; }
      }
    }
  }

  // One cross-lane reduction at the end: branchless u64-key min over the
  // 16 lanes of each half (xor masks < 16 stay inside the half).
#pragma unroll
  for (int r = 0; r < 8; ++r) {
    unsigned long long k0 = packKey(runV0[r], runI0[r]);
    unsigned long long k1 = packKey(runV1[r], runI1[r]);
#pragma unroll
    for (int off = 1; off < 16; off <<= 1) {
      const unsigned long long o0 = __shfl_xor(k0, off);
      const unsigned long long o1 = __shfl_xor(k1, off);
      if (o0 < k0) k0 = o0;
      if (o1 < k1) k1 = o1;
    }
    if (q == 0) {
      sKey[rbase + r + 8 * half][cg]      = k0;
      sKey[rbase + 16 + r + 8 * half][cg] = k1;
    }
  }
  __syncthreads();

  if (t < ROWS_PER_BLK) {
    const unsigned long long ka = sKey[t][0];
    const unsigned long long kb = sKey[t][1];
    const unsigned long long k  = (kb < ka) ? kb : ka;
    const int idx = (int)(k & 0xFFFFFFFFull);
    const int n   = blk * ROWS_PER_BLK + t;
    g_idx[n] = idx;
    out[OFF_IDX + n] = (float)idx;
    out[OFF_NEAR + (unsigned long long)n * N_EMB + idx] = 1.0f;  // fused one-hot
    atomicAdd(&g_hist[idx], 1);            // integer -> deterministic
  }
}

// ---------------------------------------------------------------------------
// Kernel 2: gather zq (NCHW) + per-block partial sum of (zq - zc)^2.
// ---------------------------------------------------------------------------
__global__ __launch_bounds__(256) void vq_gather_loss_kernel(const float* __restrict__ z,
                                                             const float* __restrict__ emb,
                                                             const int* __restrict__ g_idx,
                                                             float* __restrict__ out,
                                                             float* __restrict__ part) {
  const int t = threadIdx.x;
  float acc = 0.0f;
  const unsigned base = blockIdx.x * 2048u + t;
#pragma unroll
  for (int j = 0; j < 8; ++j) {
    const unsigned e  = base + j * 256u;     // flat NCHW index
    const unsigned b  = e >> 18;
    const unsigned c  = (e >> 10) & 255u;
    const unsigned hw = e & 1023u;
    const unsigned n  = b * 1024u + hw;
    const float zq = emb[(unsigned)g_idx[n] * EMB_DIM + c];
    const float zc = z[e];
    out[OFF_ZQ + e] = zq;                    // zq_st == zq numerically
    const float d = zq - zc;
    acc += d * d;
  }
  // fixed-order reduction: wave shuffle tree, then LDS across 8 waves
#pragma unroll
  for (int off = 16; off > 0; off >>= 1) acc += __shfl_down(acc, off);
  __shared__ float w8[8];
  if ((t & 31) == 0) w8[t >> 5] = acc;
  __syncthreads();
  if (t == 0) {
    float s = 0.0f;
#pragma unroll
    for (int i = 0; i < 8; ++i) s += w8[i];
    part[blockIdx.x] = s;
  }
}

// ---------------------------------------------------------------------------
// Kernel 3: finalize loss and perplexity (single block, fixed-order sums).
// ---------------------------------------------------------------------------
__global__ __launch_bounds__(256) void vq_finalize_kernel(const float* __restrict__ part,
                                                          const int* __restrict__ hist,
                                                          float* __restrict__ out) {
  __shared__ float red[256];
  const int t = threadIdx.x;

  float a = 0.0f;
  for (int i = t; i < NPART; i += 256) a += part[i];
  red[t] = a;
  __syncthreads();
  for (int s = 128; s > 0; s >>= 1) {
    if (t < s) red[t] += red[t + s];
    __syncthreads();
  }
  if (t == 0) out[OFF_LOSS] = (1.0f + BETA) * red[0] / 8388608.0f;
  __syncthreads();

  float p = 0.0f;
  for (int i = t; i < N_EMB; i += 256) {
    const float em = (float)hist[i] * (1.0f / (float)N_ROWS);
    p += em * logf(em + 1e-10f);
  }
  red[t] = p;
  __syncthreads();
  for (int s = 128; s > 0; s >>= 1) {
    if (t < s) red[t] += red[t + s];
    __syncthreads();
  }
  if (t == 0) out[OFF_PERP] = expf(-red[0]);
}

// ---------------------------------------------------------------------------
extern "C" void kernel_launch(void* const* d_in, const int* in_sizes, int n_in,
                              void* d_out, int out_size, void* d_ws, size_t ws_size,
                              hipStream_t stream) {
  const float* z   = (const float*)d_in[0];   // [32,256,32,32]
  const float* emb = (const float*)d_in[1];   // [1024,256]
  float* out = (float*)d_out;

  int*   ws_idx  = (int*)d_ws;
  int*   ws_hist = (int*)((char*)d_ws + WS_HIST);
  float* ws_part = (float*)((char*)d_ws + WS_PART);
  float* ws_eb2  = (float*)((char*)d_ws + WS_EB2);

  vq_init_kernel<<<2048, 256, 0, stream>>>(out, ws_hist);
  vq_eb2_kernel<<<N_EMB / 256, 256, 0, stream>>>(emb, ws_eb2);
  vq_argmin_kernel<<<N_ROWS / ROWS_PER_BLK, 256, 0, stream>>>(z, emb, ws_eb2, out,
                                                              ws_idx, ws_hist);
  vq_gather_loss_kernel<<<NPART, 256, 0, stream>>>(z, emb, ws_idx, out, ws_part);
  vq_finalize_kernel<<<1, 256, 0, stream>>>(ws_part, ws_hist, out);
}